// NNBackflowSlater2nd_21157008900518
// MI455X (gfx1250) — compile-verified
//
#include <hip/hip_runtime.h>
#include <math.h>

// ---------------- problem constants ----------------
constexpr int Bn    = 1024;
constexpr int SIZE  = 512;
constexpr int NFr   = 128;
constexpr int WIDTH = 32;
constexpr int LDA   = 129;      // padded LDS leading dim (odd -> conflict-free columns)
constexpr float PI_F = 3.14159265358979323846f;

// ---------------- workspace layout (float offsets) ----------------
constexpr int OFF_SR     = 0;                        // 512  : Re S[o][kx*4+ky]
constexpr int OFF_SI     = 512;                      // 512  : Im S
constexpr int OFF_DCR    = 1024;                     // 32   : Re DC term
constexpr int OFF_DCI    = 1056;                     // 32   : Im DC term
constexpr int OFF_U      = 1088;                     // 32   : lw . pw_W
constexpr int OFF_BASE   = 1120;                     // 32   : lb . pw_W + pw_b
constexpr int OFF_POOLED = 1152;                     // B*32 : pooled features
constexpr int OFF_R      = OFF_POOLED + Bn * WIDTH;  // B*128 ints: gathered row indices

typedef float v2f __attribute__((ext_vector_type(2)));
typedef float v8f __attribute__((ext_vector_type(8)));

// ---------------- kernel 0: batch-independent constants ----------------
__global__ __launch_bounds__(256) void precompute_kernel(
    const float* __restrict__ lw, const float* __restrict__ lb,
    const float* __restrict__ Wr, const float* __restrict__ Wi,
    const float* __restrict__ pwW, const float* __restrict__ pwb,
    float* __restrict__ ws) {
  int tid = threadIdx.x;
  for (int t = tid; t < 512; t += 256) {
    int o = t >> 4; int m = t & 15;   // m = kx*4+ky
    float sr = 0.f, si = 0.f;
#pragma unroll 4
    for (int i = 0; i < 32; ++i) {
      float w = lw[i];
      sr += w * Wr[i * 512 + o * 16 + m];
      si += w * Wi[i * 512 + o * 16 + m];
    }
    ws[OFF_SR + t] = sr;
    ws[OFF_SI + t] = si;
  }
  if (tid < 32) {
    int o = tid;
    float dr = 0.f, di = 0.f, u = 0.f, bb = 0.f;
    for (int i = 0; i < 32; ++i) {
      dr += lb[i] * Wr[i * 512 + o * 16];
      di += lb[i] * Wi[i * 512 + o * 16];
      u  += lw[i] * pwW[i * 32 + o];
      bb += lb[i] * pwW[i * 32 + o];
    }
    ws[OFF_DCR  + o] = 512.f * dr;
    ws[OFF_DCI  + o] = 512.f * di;
    ws[OFF_U    + o] = u;
    ws[OFF_BASE + o] = bb + pwb[o];
  }
}

// ---------------- kernel 1: FNO -> pooled (one block per sample) ----------------
__global__ __launch_bounds__(256) void fno_kernel(const int* __restrict__ n,
                                                  float* __restrict__ ws) {
  __shared__ float nf[512];
  __shared__ float partR[256], partI[256];
  __shared__ float Nr[16], Ni[16];
  __shared__ float lowr[512], lowi[512];
  __shared__ float pooled[32], ubuf[32], bbuf[32];
  int tid = threadIdx.x;
  int b = blockIdx.x;

  for (int t = tid; t < 512; t += 256) nf[t] = (float)n[b * 512 + t];
  if (tid < 32) { pooled[tid] = 0.f; ubuf[tid] = ws[OFF_U + tid]; bbuf[tid] = ws[OFF_BASE + tid]; }
  __syncthreads();

  // forward DFT at the 16 retained modes: Nhat[kx,ky] = sum n(x,y) e^{-2pi i(kx x/32 + ky y/16)}
  {
    int pair = tid >> 4, sub = tid & 15;
    int kx = pair >> 2, ky = pair & 3;
    float pr = 0.f, pi = 0.f;
    for (int g = sub; g < 512; g += 16) {
      int x = g >> 4, y = g & 15;
      float ang = -PI_F * ((float)(kx * x) * (1.f / 16.f) + (float)(ky * y) * (1.f / 8.f));
      float s, c; __sincosf(ang, &s, &c);
      pr += nf[g] * c; pi += nf[g] * s;
    }
    partR[tid] = pr; partI[tid] = pi;
  }
  __syncthreads();
  if ((tid & 15) == 0) {
    int pair = tid >> 4;
    float sr = 0.f, si = 0.f;
    for (int k = 0; k < 16; ++k) { sr += partR[pair * 16 + k]; si += partI[pair * 16 + k]; }
    Nr[pair] = sr; Ni[pair] = si;
  }
  __syncthreads();

  // low[o, kx, ky] = Nhat * S  (+ DC lift-bias term at mode 0)
  for (int t = tid; t < 512; t += 256) {
    int o = t >> 4, m = t & 15;
    float sr = ws[OFF_SR + t], si = ws[OFF_SI + t];
    float nr = Nr[m], ni = Ni[m];
    float lr = nr * sr - ni * si;
    float li = nr * si + ni * sr;
    if (m == 0) { lr += ws[OFF_DCR + o]; li += ws[OFF_DCI + o]; }
    lowr[t] = lr; lowi[t] = li;
  }
  __syncthreads();

  // inverse transform at each grid point + pointwise + tanh + mean-pool
  for (int pt = 0; pt < 2; ++pt) {
    int g = tid + 256 * pt;
    int x = g >> 4, y = g & 15;
    float cxr[4], cxi[4], cyc[4], cys[4];
#pragma unroll
    for (int kx = 0; kx < 4; ++kx) {
      float ang = PI_F * (float)(kx * x) * (1.f / 16.f);
      __sincosf(ang, &cxi[kx], &cxr[kx]);
    }
#pragma unroll
    for (int ky = 0; ky < 4; ++ky) {
      float ang = PI_F * (float)(ky * y) * (1.f / 8.f);
      __sincosf(ang, &cys[ky], &cyc[ky]);
    }
    float nv = nf[g];
    for (int o = 0; o < 32; ++o) {
      float hs = 0.f;
#pragma unroll
      for (int ky = 0; ky < 4; ++ky) {
        float tr = 0.f, ti = 0.f;
#pragma unroll
        for (int kx = 0; kx < 4; ++kx) {
          float lr = lowr[o * 16 + kx * 4 + ky];
          float li = lowi[o * 16 + kx * 4 + ky];
          tr += lr * cxr[kx] - li * cxi[kx];
          ti += lr * cxi[kx] + li * cxr[kx];
        }
        hs += (ky == 0) ? tr : 2.f * (tr * cyc[ky] - ti * cys[ky]);
      }
      hs *= (1.f / 512.f);
      float val = tanhf(hs + nv * ubuf[o] + bbuf[o]);
      atomicAdd(&pooled[o], val * (1.f / 512.f));   // ds_add_f32
    }
  }
  __syncthreads();
  if (tid < 32) ws[OFF_POOLED + b * 32 + tid] = pooled[tid];
}

// ---------------- kernel 2: nonzero(size=128) per sample ----------------
__global__ __launch_bounds__(256) void idx_kernel(const int* __restrict__ n,
                                                  int* __restrict__ R) {
  int b = blockIdx.x * 256 + threadIdx.x;
  if (b >= Bn) return;
  int cnt = 0;
  for (int s = 0; s < SIZE && cnt < NFr; ++s)
    if (n[b * SIZE + s] != 0) R[b * NFr + cnt++] = s;
  while (cnt < NFr) R[b * NFr + cnt++] = 0;
}

// ---------------- kernel 3: build A in LDS + blocked pivoted LU (WMMA) + logdet ----------------
__global__ __launch_bounds__(256) void det_kernel(
    const float* __restrict__ Mbase, const float* __restrict__ projW,
    const float* __restrict__ projb, const float* __restrict__ ws,
    const int* __restrict__ Rw, float* __restrict__ out) {
  extern __shared__ float As[];            // 128 x LDA
  __shared__ float pool_s[32];
  __shared__ int   R_s[128];
  __shared__ float s_sign, s_logabs;
  __shared__ int   s_prow;

  int tid = threadIdx.x;
  int b = blockIdx.x;
  if (tid < 32)  pool_s[tid] = ws[OFF_POOLED + b * 32 + tid];
  if (tid < 128) R_s[tid] = Rw[b * 128 + tid];
  if (tid == 0) { s_sign = 1.f; s_logabs = 0.f; }
  __syncthreads();

  // A[i][j] = M_base[r,j] + proj_b[r*128+j] + pooled . proj_W[:, r*128+j]
  for (int t = tid; t < 128 * 128; t += 256) {
    int i = t >> 7, j = t & 127;
    int base = R_s[i] * 128 + j;
    float acc = Mbase[base] + projb[base];
#pragma unroll
    for (int c = 0; c < 32; ++c) acc += pool_s[c] * projW[c * 65536 + base];
    As[i * LDA + j] = acc;
  }
  __syncthreads();

  // blocked right-looking LU, NB=16, partial pivoting
  for (int p = 0; p < 8; ++p) {
    int kb = 16 * p;
    // ---- panel factorization (cols kb..kb+15, rows kb..127) ----
    for (int jj = 0; jj < 16; ++jj) {
      int j = kb + jj;
      if (tid == 0) {
        int prow = j; float best = fabsf(As[j * LDA + j]);
        for (int i = j + 1; i < 128; ++i) {
          float v = fabsf(As[i * LDA + j]);
          if (v > best) { best = v; prow = i; }
        }
        s_prow = prow;
        if (prow != j) s_sign = -s_sign;
      }
      __syncthreads();
      int prow = s_prow;
      if (prow != j) {
        for (int col = tid; col < 128; col += 256) {
          float tmp = As[j * LDA + col];
          As[j * LDA + col] = As[prow * LDA + col];
          As[prow * LDA + col] = tmp;
        }
      }
      __syncthreads();
      float piv = As[j * LDA + j];
      if (tid == 0) {
        s_logabs += logf(fabsf(piv));
        if (piv < 0.f) s_sign = -s_sign;
      }
      float rp = 1.f / piv;
      for (int i = j + 1 + tid; i < 128; i += 256) As[i * LDA + j] *= rp;
      __syncthreads();
      int ncols = kb + 15 - j;
      if (ncols > 0) {
        int nrows = 127 - j;
        int tot = nrows * ncols;
        for (int t = tid; t < tot; t += 256) {
          int i = j + 1 + t / ncols;
          int c = j + 1 + t % ncols;
          As[i * LDA + c] -= As[i * LDA + j] * As[j * LDA + c];
        }
      }
      __syncthreads();
    }
    if (kb + 16 < 128) {
      // ---- triangular solve: U12 = L11^{-1} A12 ----
      int nc = 128 - (kb + 16);
      for (int jj = 1; jj < 16; ++jj) {
        int j = kb + jj;
        for (int t = tid; t < nc; t += 256) {
          int col = kb + 16 + t;
          float acc = As[j * LDA + col];
          for (int m = kb; m < j; ++m) acc -= As[j * LDA + m] * As[m * LDA + col];
          As[j * LDA + col] = acc;
        }
        __syncthreads();
      }
      // ---- trailing update A22 -= L21 * U12 via V_WMMA_F32_16X16X4_F32 ----
      // f32 WMMA has no A/B negate modifier, so accumulate in negated space:
      //   C0 = -A22 ; C += L*U (4 chained wmmas) ; A22_new = -C = A22 - L*U
      int nt = nc >> 4;              // tiles per dim = 7 - p
      int ntiles = nt * nt;
      int wave = tid >> 5;           // wave32: 8 waves per block
      int lane = tid & 31;
      int hi = lane >> 4;
      int ln = lane & 15;
      for (int t = wave; t < ntiles; t += 8) {
        int bi = kb + 16 + (t / nt) * 16;
        int bj = kb + 16 + (t % nt) * 16;
        v8f c;
#pragma unroll
        for (int v = 0; v < 8; ++v) c[v] = -As[(bi + v + 8 * hi) * LDA + bj + ln];
#pragma unroll
        for (int kk = 0; kk < 4; ++kk) {
          int k0 = kb + 4 * kk + 2 * hi;
          v2f a, bm;
          a[0]  = As[(bi + ln) * LDA + k0];        // A 16x4: M=lane, K split across 2 VGPRs
          a[1]  = As[(bi + ln) * LDA + k0 + 1];
          bm[0] = As[k0 * LDA + bj + ln];          // B 4x16: N=lane, K split across 2 VGPRs
          bm[1] = As[(k0 + 1) * LDA + bj + ln];
          c = __builtin_amdgcn_wmma_f32_16x16x4_f32(false, a, false, bm,
                                                    (short)0, c, false, false);
        }
#pragma unroll
        for (int v = 0; v < 8; ++v) As[(bi + v + 8 * hi) * LDA + bj + ln] = -c[v];
      }
      __syncthreads();
    }
  }
  __syncthreads();
  if (tid == 0) {
    out[2 * b]     = s_logabs;
    out[2 * b + 1] = (s_sign < 0.f) ? PI_F : 0.f;  // log(-1) = i*pi
  }
}

// ---------------- host launcher ----------------
extern "C" void kernel_launch(void* const* d_in, const int* in_sizes, int n_in,
                              void* d_out, int out_size, void* d_ws, size_t ws_size,
                              hipStream_t stream) {
  const int*   n       = (const int*)d_in[0];
  const float* M_base  = (const float*)d_in[1];
  const float* lift_W  = (const float*)d_in[2];
  const float* lift_b  = (const float*)d_in[3];
  const float* spec_Wr = (const float*)d_in[4];
  const float* spec_Wi = (const float*)d_in[5];
  const float* pw_W    = (const float*)d_in[6];
  const float* pw_b    = (const float*)d_in[7];
  const float* proj_W  = (const float*)d_in[8];
  const float* proj_b  = (const float*)d_in[9];
  float* out = (float*)d_out;
  float* wsf = (float*)d_ws;
  int*   wsR = (int*)(wsf + OFF_R);

  precompute_kernel<<<1, 256, 0, stream>>>(lift_W, lift_b, spec_Wr, spec_Wi, pw_W, pw_b, wsf);
  fno_kernel<<<Bn, 256, 0, stream>>>(n, wsf);
  idx_kernel<<<(Bn + 255) / 256, 256, 0, stream>>>(n, wsR);

  size_t smem = (size_t)(128 * LDA) * sizeof(float);   // 66 KB of the 320 KB WGP LDS
  (void)hipFuncSetAttribute(reinterpret_cast<const void*>(det_kernel),
                            hipFuncAttributeMaxDynamicSharedMemorySize, (int)smem);
  det_kernel<<<Bn, 256, smem, stream>>>(M_base, proj_W, proj_b, wsf, wsR, out);
}